// SGDFilterDesign_22024592294122
// MI455X (gfx1250) — compile-verified
//
#include <hip/hip_runtime.h>
#include <math.h>

typedef float v2f __attribute__((ext_vector_type(2)));
typedef float v8f __attribute__((ext_vector_type(8)));

#define N_ITERS  1000
#define LR       0.1f
#define EPS      1e-8f

// D(16x16) = A(16x4) x B(4x16) + C, fp32 WMMA (CDNA5, wave32)
static __device__ __forceinline__ v8f wmma4(v2f a, v2f b, v8f c) {
  return __builtin_amdgcn_wmma_f32_16x16x4_f32(false, a, false, b,
                                               (short)0, c, false, false);
}

// compiler-only fence: same-wave LDS ops are processed in order by HW,
// we only need to stop the compiler from reordering shared accesses.
static __device__ __forceinline__ void lds_fence() {
  asm volatile("" ::: "memory");
}

__global__ __launch_bounds__(512, 1)
void sgd_sos_design_kernel(const float* __restrict__ target_dB,
                           const float* __restrict__ sos_init,
                           float* __restrict__ out)
{
  // Coefficients kept directly in WMMA A-layout:
  //   coefA[mat][lane][2], mat 0 = b, 1 = a.
  //   lane m<16  : {c[m][k0+0], c[m][k0+1]}
  //   lane 16+m  : {c[m][k0+2], 0}
  __shared__ float coefA[2][32][2];
  __shared__ float grad[16][8];            // [section][k], k<6 used
  // per-wave transpose buffers: [wave][mat(X/Y)][section][freq(+pad to 18)]
  __shared__ float tpose[16][2][16][18];

  const int tid  = threadIdx.x;
  const int lane = tid & 31;
  const int wv   = tid >> 5;     // wave id: 0..15, owns freqs [wv*32, wv*32+32)
  const int m    = lane & 15;    // column / row index inside 16x16 tiles
  const int hi   = lane >> 4;    // lane half

  // ---- init: scatter sos_init into A-layout, zero grads and K=3 slots ----
  if (tid < 96) {
    const int s = tid / 6, k = tid % 6;
    const int mat = k / 3, kk = k % 3;
    const int row = s + ((kk == 2) ? 16 : 0);
    const int col = (kk == 2) ? 0 : kk;
    coefA[mat][row][col] = sos_init[tid];
  }
  if (tid < 16)  { coefA[0][tid + 16][1] = 0.f; coefA[1][tid + 16][1] = 0.f; }
  if (tid < 128) grad[tid >> 3][tid & 7] = 0.f;

  // ---------------- per-thread constants (computed once) ----------------
  const float PI_OVER = 3.14159265358979323846f / 511.0f;  // linspace(0,pi,512)
  float tgt[2];
  v2f Pre[2], Pim[2];        // forward B-matrices: powers of z^-1 (re / im)
  v2f gBc[2][4], gBs[2][4];  // gradient B-matrices: cos(k*w) / -sin(k*w)
  #pragma unroll
  for (int t = 0; t < 2; ++t) {
    const int f = wv * 32 + t * 16 + m;
    const float w = PI_OVER * (float)f;
    tgt[t] = target_dB[f];
    // B layout (4x16 fp32): vgpr0 <-> row k=2*hi, vgpr1 <-> row k=2*hi+1, col=m
    Pre[t][0] = hi ?  cosf(2.f * w) : 1.0f;
    Pre[t][1] = hi ?  0.f           : cosf(w);
    Pim[t][0] = hi ? -sinf(2.f * w) : 0.f;
    Pim[t][1] = hi ?  0.f           : -sinf(w);
    #pragma unroll
    for (int c = 0; c < 4; ++c) {
      const int f0 = wv * 32 + t * 16 + 4 * c + 2 * hi;  // freq of row kk=2*hi
      const float w0 = PI_OVER * (float)f0;
      const float w1 = PI_OVER * (float)(f0 + 1);
      const float jf = (float)m;        // output column j = coefficient index k
      const bool valid = (m < 3);       // only k = 0..2 meaningful
      gBc[t][c][0] = valid ?  cosf(jf * w0) : 0.f;
      gBc[t][c][1] = valid ?  cosf(jf * w1) : 0.f;
      gBs[t][c][0] = valid ? -sinf(jf * w0) : 0.f;
      gBs[t][c][1] = valid ? -sinf(jf * w1) : 0.f;
    }
  }
  __syncthreads();

  const float GCONST = 0.0339292578f;  // (2/512)*(20/ln10)
  const float DB2    = 6.02059991f;    // 20/log2(10)

  for (int it = 0; it < N_ITERS; ++it) {
    // unconditional, conflict-free A-matrix loads (ds_load_b64 per matrix)
    v2f Ab = *(const v2f*)&coefA[0][lane][0];
    v2f Aa = *(const v2f*)&coefA[1][lane][0];

    v8f Cb = {};  // grad accumulator for b (cols 0..2 valid)
    v8f Ca = {};  // grad accumulator for a

    #pragma unroll
    for (int t = 0; t < 2; ++t) {
      // ---- forward: num/den real & imag as 16(sec) x 16(freq) tiles ----
      v8f z = {};
      v8f nr = wmma4(Ab, Pre[t], z);
      v8f ni = wmma4(Ab, Pim[t], z);
      v8f dr = wmma4(Aa, Pre[t], z);
      v8f di = wmma4(Aa, Pim[t], z);

      // squared magnitudes per section (vector form -> v_pk_* friendly)
      v8f n2 = nr * nr + ni * ni;
      v8f d2 = dr * dr + di * di;
      v8f rn2, rd2;
      #pragma unroll
      for (int v = 0; v < 8; ++v) {
        rn2[v] = __builtin_amdgcn_rcpf(n2[v]);
        rd2[v] = __builtin_amdgcn_rcpf(d2[v]);
      }

      // |H|^2 = prod over all 16 sections of n2/d2 (8 here + partner half)
      v8f ratio = n2 * rd2;
      float p = ((ratio[0] * ratio[1]) * (ratio[2] * ratio[3])) *
                ((ratio[4] * ratio[5]) * (ratio[6] * ratio[7]));
      p *= __shfl_xor(p, 16, 32);                 // combine section halves
      const float absH = __builtin_amdgcn_sqrtf(p);
      const float mag  = DB2 * __builtin_amdgcn_logf(absH + EPS);
      const float g    = GCONST * (mag - tgt[t]) * absH *
                         __builtin_amdgcn_rcpf(absH + EPS);

      // ---- numerator grad: X = (g/|num|^2)*num_re, Y = ... *num_im ----
      {
        v8f gn = g * rn2;
        v8f X = gn * nr, Y = gn * ni;
        #pragma unroll
        for (int v = 0; v < 8; ++v) {
          tpose[wv][0][8 * hi + v][m] = X[v];
          tpose[wv][1][8 * hi + v][m] = Y[v];
        }
      }
      lds_fence();
      #pragma unroll
      for (int c = 0; c < 4; ++c) {
        v2f ax = *(const v2f*)&tpose[wv][0][m][4 * c + 2 * hi];
        v2f ay = *(const v2f*)&tpose[wv][1][m][4 * c + 2 * hi];
        Cb = wmma4(ax, gBc[t][c], Cb);
        Cb = wmma4(ay, gBs[t][c], Cb);
      }
      lds_fence();

      // ---- denominator grad (note the minus sign) ----
      {
        v8f gd = (-g) * rd2;
        v8f X = gd * dr, Y = gd * di;
        #pragma unroll
        for (int v = 0; v < 8; ++v) {
          tpose[wv][0][8 * hi + v][m] = X[v];
          tpose[wv][1][8 * hi + v][m] = Y[v];
        }
      }
      lds_fence();
      #pragma unroll
      for (int c = 0; c < 4; ++c) {
        v2f ax = *(const v2f*)&tpose[wv][0][m][4 * c + 2 * hi];
        v2f ay = *(const v2f*)&tpose[wv][1][m][4 * c + 2 * hi];
        Ca = wmma4(ax, gBc[t][c], Ca);
        Ca = wmma4(ay, gBs[t][c], Ca);
      }
      lds_fence();
    }

    // ---- scatter per-wave grads: C element (s,j) lives in lane j+16*(s>=8),
    // vgpr s%8; only j<3 meaningful ----
    if (m < 3) {
      #pragma unroll
      for (int v = 0; v < 8; ++v) {
        atomicAdd(&grad[8 * hi + v][m],     Cb[v]);   // ds_add_f32
        atomicAdd(&grad[8 * hi + v][3 + m], Ca[v]);
      }
    }
    __syncthreads();

    // ---- SGD update (in A-layout) + gradient reset ----
    if (tid < 96) {
      const int s = tid / 6, k = tid % 6;
      const int mat = k / 3, kk = k % 3;
      const int row = s + ((kk == 2) ? 16 : 0);
      const int col = (kk == 2) ? 0 : kk;
      coefA[mat][row][col] -= LR * grad[s][k];
      grad[s][k] = 0.f;
    }
    __syncthreads();
  }

  if (tid < 96) {
    const int s = tid / 6, k = tid % 6;
    const int mat = k / 3, kk = k % 3;
    const int row = s + ((kk == 2) ? 16 : 0);
    const int col = (kk == 2) ? 0 : kk;
    out[tid] = coefA[mat][row][col];
  }
}

extern "C" void kernel_launch(void* const* d_in, const int* in_sizes, int n_in,
                              void* d_out, int out_size, void* d_ws, size_t ws_size,
                              hipStream_t stream) {
  const float* target = (const float*)d_in[0];  // target_dB: 512 fp32
  const float* sos    = (const float*)d_in[1];  // sos_init: 96 fp32
  float* out = (float*)d_out;                   // 96 fp32
  (void)in_sizes; (void)n_in; (void)out_size; (void)d_ws; (void)ws_size;
  sgd_sos_design_kernel<<<dim3(1), dim3(512), 0, stream>>>(target, sos, out);
}